// DialogActLabeller_44100724195687
// MI455X (gfx1250) — compile-verified
//
#include <hip/hip_runtime.h>

typedef __attribute__((ext_vector_type(2))) float v2f;
typedef __attribute__((ext_vector_type(4))) float f4;
typedef __attribute__((ext_vector_type(8))) float v8f;

// ---------------------------------------------------------------------------
// Kernel 1: segment row-sum over enc_output.
// One block per (batch, segment); 256 threads each own one float4 (4 dims) of
// D_IN=1024 and stream the segment's rows. This kernel reads the 512MB input
// exactly once -> it is the memory-bound dominant cost (~22us at 23.3 TB/s).
// Non-temporal loads: enc is single-use, keep L2 for W / seg_sums instead.
// ---------------------------------------------------------------------------
__global__ void __launch_bounds__(256) seg_reduce_kernel(
    const float* __restrict__ enc, const int* __restrict__ cls_pos,
    const int* __restrict__ last_sep, float* __restrict__ seg_sums,
    float* __restrict__ counts, int S, int D, int N) {
  int seg = blockIdx.x;           // b * N + s
  int b = seg / N;
  int s = seg - b * N;
  int start = cls_pos[seg];
  int end = (s < N - 1) ? cls_pos[seg + 1] : (last_sep[b] + 1);
  if (s == N - 1 && end <= start) end = S;   // torch semantics for last segment

  int d4 = D >> 2;                 // float4s per row
  int t = threadIdx.x;             // 0 .. d4-1
  const f4* base = (const f4*)(enc + (size_t)b * S * D) + t;

  f4 a0 = {0.f, 0.f, 0.f, 0.f};
  f4 a1 = {0.f, 0.f, 0.f, 0.f};
  int r = start;
  for (; r + 2 <= end; r += 2) {
    f4 v0 = __builtin_nontemporal_load(&base[(size_t)r * d4]);
    f4 v1 = __builtin_nontemporal_load(&base[(size_t)(r + 1) * d4]);
    a0 += v0;
    a1 += v1;
  }
  if (r < end) {
    a0 += __builtin_nontemporal_load(&base[(size_t)r * d4]);
  }
  f4 acc = a0 + a1;
  *((f4*)(seg_sums + (size_t)seg * D) + t) = acc;
  if (t == 0) counts[seg] = (float)(end - start);
}

// ---------------------------------------------------------------------------
// Kernel 2: C[M,Nc] = A[M,K] @ W[K,Nc] + counts[m] * bias[n]
// One wave per 16x16 C tile using V_WMMA_F32_16X16X4_F32 (full f32: GEMM is
// only ~1 GFLOP after the segment-sum reduction, so precision > rate here).
// Layouts per CDNA5 ISA 7.12.2 (32-bit A 16x4, mirrored B 4x16, f32 C/D).
// ---------------------------------------------------------------------------
__global__ void __launch_bounds__(256) wmma_gemm_bias_kernel(
    const float* __restrict__ A, const float* __restrict__ Wm,
    const float* __restrict__ bias, const float* __restrict__ counts,
    float* __restrict__ C, int M, int K, int Nc) {
  int wave = (int)((blockIdx.x * blockDim.x + threadIdx.x) >> 5);
  int lane = (int)(threadIdx.x & 31);
  int ntiles = Nc >> 4;
  int mt = wave / ntiles;
  int nt = wave - mt * ntiles;
  if (mt * 16 >= M) return;        // wave-uniform: EXEC stays all-1s for WMMA

  int half = lane >> 4;            // lane group: 0 => K/M low half, 1 => high
  int l = lane & 15;
  const float* Arow = A + (size_t)(mt * 16 + l) * K;   // A row for this lane
  int bcol = nt * 16 + l;                              // B/C column

  v8f acc = {};
  for (int k = 0; k < K; k += 4) {
    int ka = k + (half << 1);
    // A frag: lane<16 holds {K=k, k+1}; lane>=16 holds {K=k+2, k+3}
    v2f a = *(const v2f*)(Arow + ka);
    // B frag mirrors A: VGPR0 rows k / k+2 per half, VGPR1 rows k+1 / k+3
    v2f bv;
    bv.x = Wm[(size_t)ka * Nc + bcol];
    bv.y = Wm[(size_t)(ka + 1) * Nc + bcol];
    acc = __builtin_amdgcn_wmma_f32_16x16x4_f32(
        /*neg_a=*/false, a, /*neg_b=*/false, bv,
        /*c_mod=*/(short)0, acc, /*reuse_a=*/false, /*reuse_b=*/false);
  }

  // D layout: VGPR r, lanes 0-15 => row r, lanes 16-31 => row r+8; col = l
  float bb = bias[bcol];
  int rowbase = mt * 16 + (half << 3);
#pragma unroll
  for (int r = 0; r < 8; ++r) {
    int row = rowbase + r;
    C[(size_t)row * Nc + bcol] = acc[r] + counts[row] * bb;
  }
}

// ---------------------------------------------------------------------------
// Kernel 3: row-wise log_softmax over Nc (=256) classes. One block per row.
// ---------------------------------------------------------------------------
__global__ void logsoftmax_kernel(const float* __restrict__ logits,
                                  float* __restrict__ out, int Nc) {
  int row = blockIdx.x;
  int t = threadIdx.x;
  __shared__ float red[1024];

  float v = logits[(size_t)row * Nc + t];
  red[t] = v;
  __syncthreads();
  for (int off = (int)(blockDim.x >> 1); off > 0; off >>= 1) {
    if (t < off) red[t] = fmaxf(red[t], red[t + off]);
    __syncthreads();
  }
  float m = red[0];
  __syncthreads();
  red[t] = expf(v - m);
  __syncthreads();
  for (int off = (int)(blockDim.x >> 1); off > 0; off >>= 1) {
    if (t < off) red[t] += red[t + off];
    __syncthreads();
  }
  float lse = logf(red[0]);
  out[(size_t)row * Nc + t] = v - m - lse;
}

// ---------------------------------------------------------------------------
extern "C" void kernel_launch(void* const* d_in, const int* in_sizes, int n_in,
                              void* d_out, int out_size, void* d_ws, size_t ws_size,
                              hipStream_t stream) {
  const float* enc      = (const float*)d_in[0];  // [B,S,D_IN] f32
  const float* W        = (const float*)d_in[1];  // [D_IN,D_OUT] f32
  const float* bias     = (const float*)d_in[2];  // [D_OUT] f32
  // d_in[3] = max_num_sent scalar (unused; derived from shapes below)
  const int*   cls_pos  = (const int*)d_in[4];    // [B,N] i32
  const int*   last_sep = (const int*)d_in[5];    // [B] i32

  const int B     = in_sizes[5];
  const int D_OUT = in_sizes[2];
  const int D_IN  = in_sizes[1] / D_OUT;
  const int N     = in_sizes[4] / B;
  const int S     = in_sizes[0] / (B * D_IN);
  const int M     = B * N;                         // 2048 segment rows

  // Workspace layout: seg_sums [M,D_IN] | counts [M] | logits [M,D_OUT]
  float* seg_sums = (float*)d_ws;
  float* counts   = seg_sums + (size_t)M * D_IN;
  float* logits   = counts + M;

  // 1) segment row-sums (memory-bound phase: 512MB streamed once)
  seg_reduce_kernel<<<M, D_IN / 4, 0, stream>>>(
      enc, cls_pos, last_sep, seg_sums, counts, S, D_IN, N);

  // 2) f32 WMMA GEMM + per-row-count bias
  int waves  = (M / 16) * (D_OUT / 16);
  int blocks = (waves + 7) / 8;                    // 8 waves (256 thr) / block
  wmma_gemm_bias_kernel<<<blocks, 256, 0, stream>>>(
      seg_sums, W, bias, counts, logits, M, D_IN, D_OUT);

  // 3) log-softmax over classes
  logsoftmax_kernel<<<M, D_OUT, 0, stream>>>(logits, (float*)d_out, D_OUT);
}